// BasePolicyNetwork_41102837022780
// MI455X (gfx1250) — compile-verified
//
#include <hip/hip_runtime.h>
#include <hip/hip_bf16.h>
#include <stdint.h>

// Problem sizes (fixed by the reference).
#define N_ROWS   65536
#define K_IN     513      // 1 bias column + 512 weight columns
#define K_W      512
#define OUT_C    1024
#define DECAY    1e-4f

#define CHUNK_K   16                    // weight columns staged per TDM chunk
#define N_CHUNKS  (K_W / CHUNK_K)       // 32
#define LDS_COL_STRIDE 17               // 16 data words + 1 pad word (bank-conflict-free)
#define LDS_CHUNK_WORDS (OUT_C * LDS_COL_STRIDE)   // 17408 words = 69632 B

typedef float v2f __attribute__((ext_vector_type(2)));
typedef float v8f __attribute__((ext_vector_type(8)));
typedef unsigned int uint32x4 __attribute__((ext_vector_type(4)));
typedef int int32x8 __attribute__((ext_vector_type(8)));
typedef int int32x4 __attribute__((ext_vector_type(4)));

#if __has_builtin(__builtin_amdgcn_tensor_load_to_lds)
#define USE_TDM 1
#else
#define USE_TDM 0
#endif

// ---------------------------------------------------------------------------
// Deterministic counter-based RNG (same element -> same stream every call).
// ---------------------------------------------------------------------------
__device__ __forceinline__ float u01(uint64_t& s) {
    s ^= s >> 12; s ^= s << 25; s ^= s >> 27;
    uint64_t r = s * 0x2545F4914F6CDD1DULL;
    return ((float)((r >> 40) & 0xFFFFFFu) + 0.5f) * (1.0f / 16777216.0f);
}

__device__ __forceinline__ float normal01(uint64_t& s) {
    float u1 = u01(s);
    float u2 = u01(s);
    return sqrtf(-2.0f * __logf(u1)) * __cosf(6.28318530718f * u2);
}

// Marsaglia-Tsang Gamma(alpha,1) with alpha<1 boost. noinline: one code copy.
__device__ __attribute__((noinline)) float gamma_rsample(float alpha, uint64_t s) {
    float boost = 1.0f;
    if (alpha < 1.0f) {
        boost = __powf(u01(s), 1.0f / alpha);
        alpha += 1.0f;
    }
    const float d = alpha - (1.0f / 3.0f);
    const float c = __frsqrt_rn(9.0f * d);
    float v = 1.0f;
    #pragma unroll 1
    for (int it = 0; it < 8; ++it) {
        float x = normal01(s);
        float t = 1.0f + c * x;
        if (t <= 0.0f) continue;
        v = t * t * t;
        float u = u01(s);
        if (__logf(u) < 0.5f * x * x + d * (1.0f - v + __logf(v))) break;
    }
    return boost * d * v;
}

__device__ __forceinline__ void atomicMaxF(float* addr, float val) {
    unsigned int* ua = (unsigned int*)addr;
    unsigned int old = *ua;
    while (true) {
        if (__uint_as_float(old) >= val) break;
        unsigned int assumed = old;
        old = atomicCAS(ua, assumed, __float_as_uint(val));
        if (old == assumed) break;
    }
}

#if USE_TDM
// ---------------------------------------------------------------------------
// Tensor Data Mover: DMA a 16(K) x 1024(N) tile of conv_w into LDS.
// conv_w is [1024 x 513] row-major; tile dim0 (x) = 16 contiguous K values,
// tile dim1 (y) = 1024 rows. pad_interval=16 DWORDs / pad_amount=1 DWORD gives
// a 17-word per-column LDS stride (17 coprime 64 banks -> conflict-free reads).
// ---------------------------------------------------------------------------
__device__ __forceinline__ void tdm_load_chunk(const float* conv_w, int chunk,
                                               const float* ldsbuf) {
    uint64_t gaddr = (uint64_t)(uintptr_t)conv_w + 4ull * (uint64_t)(1 + CHUNK_K * chunk);
    uint32_t laddr = (uint32_t)(uintptr_t)ldsbuf;       // LDS byte offset
    uint32x4 g0;
    g0[0] = 1u;                                          // count=1 (valid user D#)
    g0[1] = laddr;                                       // lds_addr
    g0[2] = (uint32_t)(gaddr & 0xFFFFFFFFu);             // global_addr[31:0]
    g0[3] = (uint32_t)((gaddr >> 32) & 0x1FFFFFFu)       // global_addr[56:32]
            | (2u << 30);                                // type = 2 ("image")
    int32x8 g1;
    g1[0] = (2 << 16)        // data_size = 4 bytes
          | (1 << 20)        // pad_enable
          | (3 << 22)        // pad_interval code 3 = 16 DWORDs
          | (0 << 25);       // pad_amount  code 0 = 1 DWORD
    g1[1] = (int)(513u  << 16);   // tensor_dim0[15:0]  (bits 63:48)
    g1[2] = (int)(1024u << 16);   // tensor_dim1[15:0]  (bits 95:80)
    g1[3] = (int)(16u   << 16);   // tile_dim0 = 16     (bits 127:112)
    g1[4] = 1024;                 // tile_dim1 = 1024   (bits 143:128)
    g1[5] = 513;                  // tensor_dim0_stride[31:0]
    g1[6] = 0;
    g1[7] = 0;
    int32x4 z4 = {0, 0, 0, 0};            // groups 2/3 unused (tile_dim2 == 0)
    int32x8 z8 = {0, 0, 0, 0, 0, 0, 0, 0};
    __builtin_amdgcn_tensor_load_to_lds(g0, g1, z4, z4, z8, 0);
}
#endif

// ---------------------------------------------------------------------------
// Fused: FP32 WMMA GEMM (32x1024 tile per workgroup, TDM-staged B) + softmax
// + Dirichlet rsample + row normalize + concen^2 regularizer.
// ---------------------------------------------------------------------------
__global__ __launch_bounds__(256) void fused_dirichlet_kernel(
    const float* __restrict__ bias,    // [N]
    const float* __restrict__ weight,  // [N,512]
    const float* __restrict__ prior,   // [1,1024]
    const float* __restrict__ conv_w,  // [1024,513] row-major
    const float* __restrict__ conv_b,  // [1024]
    float* __restrict__ out)           // [N*1024 hierarchy] + [1 reg]
{
    __shared__ __align__(16) float bstage[2][LDS_CHUNK_WORDS];  // ~136 KB
    __shared__ float rowMax[32];
    __shared__ float rowSumE[32];
    __shared__ float rowSumG[32];

    const int tid   = threadIdx.x;
    const int wave  = tid >> 5;
    const int lane  = tid & 31;
    const int lmod  = lane & 15;   // N (or M for A) position within half-wave
    const int lgrp  = lane >> 4;   // K-pair selector / C row-half selector
    const int mHalf = wave >> 2;   // which 16-row half of the 32-row block
    const int nQuad = wave & 3;    // which 256-col quarter of 1024 cols
    const int rowBase = blockIdx.x * 32 + 16 * mHalf;
    const int nBase   = nQuad * 256;

    if (tid < 32) { rowMax[tid] = -3.0e38f; rowSumE[tid] = 0.0f; rowSumG[tid] = 0.0f; }

    // 16 C-tiles of 16x16 per wave: 128 accumulator VGPRs/lane.
    v8f acc[16];
    #pragma unroll
    for (int t = 0; t < 16; ++t)
        #pragma unroll
        for (int i = 0; i < 8; ++i) acc[t][i] = 0.0f;

    // A fragment rows: lane holds M = lane%16, K-pair = 2*(lane/16) (+vgpr).
    const float* arow = weight + (size_t)(rowBase + lmod) * K_W;

    // ---- stage chunk 0 ----
#if USE_TDM
    if (wave == 0) tdm_load_chunk(conv_w, 0, &bstage[0][0]);
#else
    for (int e = tid; e < OUT_C * CHUNK_K; e += 256) {
        int col = e >> 4, x = e & 15;
        bstage[0][col * LDS_COL_STRIDE + x] = conv_w[(size_t)col * K_IN + 1 + x];
    }
#endif
    __syncthreads();   // also covers rowMax/rowSum init

    // ---- main K loop: 32 chunks x (4 WMMA K-steps of K=4) ----
    for (int ch = 0; ch < N_CHUNKS; ++ch) {
        const float* buf = &bstage[ch & 1][0];
#if USE_TDM
        if (wave == 0) __builtin_amdgcn_s_wait_tensorcnt(0);
#endif
        __syncthreads();           // chunk ch staged; previous buffer free
        if (ch + 1 < N_CHUNKS) {
#if USE_TDM
            if (wave == 0) tdm_load_chunk(conv_w, ch + 1, &bstage[(ch + 1) & 1][0]);
#else
            for (int e = tid; e < OUT_C * CHUNK_K; e += 256) {
                int col = e >> 4, x = e & 15;
                bstage[(ch + 1) & 1][col * LDS_COL_STRIDE + x] =
                    conv_w[(size_t)col * K_IN + 1 + CHUNK_K * (ch + 1) + x];
            }
#endif
        }

        // Preload all 4 A fragments of this chunk (overlap loads with WMMA).
        const int kwBase = ch * CHUNK_K;
        __builtin_prefetch(arow + kwBase + CHUNK_K, 0, 1);   // global_prefetch_b8
        float2 a4[4];
        #pragma unroll
        for (int kk = 0; kk < 4; ++kk)
            a4[kk] = *(const float2*)(arow + kwBase + kk * 4 + 2 * lgrp);

        #pragma unroll
        for (int kk = 0; kk < 4; ++kk) {
            v2f a; a.x = a4[kk].x; a.y = a4[kk].y;
            const int koff = kk * 4 + 2 * lgrp;

            // Preload ALL 16 B fragments first so the LDS loads pipeline
            // (clause of ds_load_2addr_b32) instead of load->wait->wmma.
            v2f bf[16];
            #pragma unroll
            for (int t = 0; t < 16; ++t) {
                const float* bp = buf + (size_t)(nBase + 16 * t + lmod) * LDS_COL_STRIDE + koff;
                bf[t].x = bp[0];
                bf[t].y = bp[1];
            }
            #pragma unroll
            for (int t = 0; t < 16; ++t) {
                acc[t] = __builtin_amdgcn_wmma_f32_16x16x4_f32(
                    false, a, false, bf[t], (short)0, acc[t], false, false);
            }
        }
    }

    // Epilogue. C layout: lane -> (M = r + 8*lgrp, N = lmod) for VGPR r.
    float biasv[8];
    #pragma unroll
    for (int r = 0; r < 8; ++r) biasv[r] = bias[rowBase + r + 8 * lgrp];

    float ssq = 0.0f;                                    // for reg
    #pragma unroll
    for (int t = 0; t < 16; ++t) {
        const int col = nBase + 16 * t + lmod;
        const float w0 = conv_w[(size_t)col * K_IN];     // bias (state col 0) weight
        const float cb = conv_b[col];
        const float pr = prior[col];
        #pragma unroll
        for (int r = 0; r < 8; ++r) {
            float c = acc[t][r] + biasv[r] * w0 + cb;    // concen (pre-prior)
            ssq += c * c;
            acc[t][r] = c + pr;                          // softmax input
        }
    }
    // reg = mean(concen^2) * DECAY ; one atomic per wave.
    #pragma unroll
    for (int off = 16; off > 0; off >>= 1) ssq += __shfl_xor(ssq, off, 32);
    if (lane == 0)
        atomicAdd(out + (size_t)N_ROWS * OUT_C,
                  ssq * (DECAY / ((float)N_ROWS * (float)OUT_C)));

    // ---- row max (stable softmax) ----
    #pragma unroll
    for (int r = 0; r < 8; ++r) {
        float m = -3.0e38f;
        #pragma unroll
        for (int t = 0; t < 16; ++t) m = fmaxf(m, acc[t][r]);
        #pragma unroll
        for (int off = 1; off < 16; off <<= 1) m = fmaxf(m, __shfl_xor(m, off, 32));
        if (lmod == 0) atomicMaxF(&rowMax[16 * mHalf + 8 * lgrp + r], m);
    }
    __syncthreads();

    // ---- exp + row sum ----
    #pragma unroll
    for (int r = 0; r < 8; ++r) {
        const float rmx = rowMax[16 * mHalf + 8 * lgrp + r];
        float s = 0.0f;
        #pragma unroll
        for (int t = 0; t < 16; ++t) {
            float e = __expf(acc[t][r] - rmx);
            acc[t][r] = e;
            s += e;
        }
        #pragma unroll
        for (int off = 1; off < 16; off <<= 1) s += __shfl_xor(s, off, 32);
        if (lmod == 0) atomicAdd(&rowSumE[16 * mHalf + 8 * lgrp + r], s);
    }
    __syncthreads();

    // ---- Gamma(alpha) rsample + row sum of samples ----
    #pragma unroll
    for (int r = 0; r < 8; ++r) {
        const int row = rowBase + r + 8 * lgrp;
        const float inv = 1.0f / rowSumE[16 * mHalf + 8 * lgrp + r];
        float gs = 0.0f;
        #pragma unroll
        for (int t = 0; t < 16; ++t) {
            const int col = nBase + 16 * t + lmod;
            const float alpha = acc[t][r] * inv;         // new_concen
            uint64_t s = (uint64_t)row * (uint64_t)OUT_C + (uint64_t)col;
            s = (s ^ 0x9E3779B97F4A7C15ULL) * 0xBF58476D1CE4E5B9ULL + 42ULL;
            s ^= s >> 31;  s |= 1ULL;
            float g = gamma_rsample(alpha, s);
            acc[t][r] = g;
            gs += g;
        }
        #pragma unroll
        for (int off = 1; off < 16; off <<= 1) gs += __shfl_xor(gs, off, 32);
        if (lmod == 0) atomicAdd(&rowSumG[16 * mHalf + 8 * lgrp + r], gs);
    }
    __syncthreads();

    // ---- normalize + store hierarchy ----
    #pragma unroll
    for (int r = 0; r < 8; ++r) {
        const int row = rowBase + r + 8 * lgrp;
        const float invg = 1.0f / rowSumG[16 * mHalf + 8 * lgrp + r];
        #pragma unroll
        for (int t = 0; t < 16; ++t) {
            const int col = nBase + 16 * t + lmod;
            out[(size_t)row * OUT_C + col] = acc[t][r] * invg;
        }
    }
}

__global__ void init_reg_kernel(float* out) {
    if (threadIdx.x == 0 && blockIdx.x == 0)
        out[(size_t)N_ROWS * OUT_C] = 0.0f;
}

extern "C" void kernel_launch(void* const* d_in, const int* in_sizes, int n_in,
                              void* d_out, int out_size, void* d_ws, size_t ws_size,
                              hipStream_t stream) {
    const float* bias   = (const float*)d_in[0];
    const float* weight = (const float*)d_in[1];
    const float* prior  = (const float*)d_in[2];
    const float* conv_w = (const float*)d_in[3];
    const float* conv_b = (const float*)d_in[4];
    float* out = (float*)d_out;

    init_reg_kernel<<<1, 32, 0, stream>>>(out);
    fused_dirichlet_kernel<<<dim3(N_ROWS / 32), 256, 0, stream>>>(
        bias, weight, prior, conv_w, conv_b, out);
}